// PriorBCE_48515950576388
// MI455X (gfx1250) — compile-verified
//
#include <hip/hip_runtime.h>
#include <math.h>

// ---------------------------------------------------------------------------
// MI455X notes:
//  - Workload is bandwidth-bound (~689 MB @ 23.3 TB/s ≈ 30 us floor), so the
//    design goal is one pass over every tensor with b128 loads.
//  - Online logsumexp (m,s running pair) removes the need for a separate
//    max pass over the 3 x 164 MB logits tensors.
//  - Cross-lane sums use V_WMMA_F32_16X16X4_F32 with a ones B-matrix:
//    A vgpr0 = lane value, vgpr1 = 0  ->  D[i][j] = v_i + v_{i+16}; summing a
//    lane's 8 D registers + one shfl_xor(16) yields the exact 32-lane sum.
//    wave32, EXEC all-1s at every call site (full 256-thread blocks).
// ---------------------------------------------------------------------------

typedef __attribute__((ext_vector_type(2))) float v2f;
typedef __attribute__((ext_vector_type(8))) float v8f;

#define BB 4096
#define VV 10000
#define DD 512
#define V4 (VV / 4)           // 2500 float4 per row
#define GAUSS_N4 ((BB * DD) / 4)   // 524288 float4 total
#define K2_BLOCKS 2048        // 2048 * 256 threads * 1 float4 = 524288

__device__ __forceinline__ float wave_sum32(float v) {
  // A: 16x4 f32 (2 VGPRs). Lanes 0-15: vgpr0->K0, vgpr1->K1; lanes 16-31: K2,K3.
  v2f a; a[0] = v;    a[1] = 0.0f;
  v2f b; b[0] = 1.0f; b[1] = 1.0f;          // B = ones(4x16): layout-agnostic
  v8f c = {0.f, 0.f, 0.f, 0.f, 0.f, 0.f, 0.f, 0.f};
  c = __builtin_amdgcn_wmma_f32_16x16x4_f32(false, a, false, b, (short)0, c,
                                            false, false);
  // lane t<16 now holds rows 0..7 of D (each = v_r + v_{r+16}); lanes >=16 rows 8..15
  float p = c[0] + c[1] + c[2] + c[3] + c[4] + c[5] + c[6] + c[7];
  p += __shfl_xor(p, 16, 32);
  return p;  // full 32-lane sum on every lane
}

__device__ __forceinline__ float block_sum256(float v, float* buf8) {
  const int lane = threadIdx.x & 31, wv = threadIdx.x >> 5;
  float w = wave_sum32(v);
  if (lane == 0) buf8[wv] = w;
  __syncthreads();
  float r = 0.f;
  if (threadIdx.x == 0) {
#pragma unroll
    for (int i = 0; i < 8; ++i) r += buf8[i];
  }
  __syncthreads();   // buffer reusable by caller
  return r;          // valid on thread 0 only
}

// ---------------------------------------------------------------------------
// Kernel 1: one block per row. Single streaming pass over recon_x, logits_rec,
// logits_text, x. Produces per-row partial  dot(L,x) - (max+lse)*sum(x)  for
// each of the 3 logits tensors into ws0[t*4096 + row].
// ---------------------------------------------------------------------------
__global__ __launch_bounds__(256) void bce_row_kernel(
    const float* __restrict__ reconX, const float* __restrict__ logitsRec,
    const float* __restrict__ logitsText, const float* __restrict__ X,
    float* __restrict__ ws0) {
  const int row = blockIdx.x;
  const int tid = threadIdx.x;
  const size_t off = (size_t)row * (size_t)VV;   // row base: 40000 B, 16B aligned
  const float4* A4 = reinterpret_cast<const float4*>(reconX + off);
  const float4* B4 = reinterpret_cast<const float4*>(logitsRec + off);
  const float4* C4 = reinterpret_cast<const float4*>(logitsText + off);
  const float4* X4 = reinterpret_cast<const float4*>(X + off);

  float m[3] = {-INFINITY, -INFINITY, -INFINITY};
  float s[3] = {0.f, 0.f, 0.f};
  float d[3] = {0.f, 0.f, 0.f};
  float sx = 0.f;

  for (int i = tid; i < V4; i += 256) {
    float4 a = A4[i], b = B4[i], c = C4[i], x = X4[i];
    float za[4] = {a.x, a.y, a.z, a.w};
    float zb[4] = {b.x, b.y, b.z, b.w};
    float zc[4] = {c.x, c.y, c.z, c.w};
    float zx[4] = {x.x, x.y, x.z, x.w};
#pragma unroll
    for (int j = 0; j < 4; ++j) {
      const float xv = zx[j];
      {
        float z = za[j]; float mn = fmaxf(m[0], z);
        s[0] = fmaf(s[0], __expf(m[0] - mn), __expf(z - mn)); m[0] = mn;
        d[0] = fmaf(z, xv, d[0]);
      }
      {
        float z = zb[j]; float mn = fmaxf(m[1], z);
        s[1] = fmaf(s[1], __expf(m[1] - mn), __expf(z - mn)); m[1] = mn;
        d[1] = fmaf(z, xv, d[1]);
      }
      {
        float z = zc[j]; float mn = fmaxf(m[2], z);
        s[2] = fmaf(s[2], __expf(m[2] - mn), __expf(z - mn)); m[2] = mn;
        d[2] = fmaf(z, xv, d[2]);
      }
      sx += xv;
    }
  }

  __shared__ float Lm[3][8], Ls[3][8], Ld[3][8], Lsx[8];
  const int lane = tid & 31, wv = tid >> 5;
#pragma unroll
  for (int t = 0; t < 3; ++t) {
    float mm = m[t];
#pragma unroll
    for (int o = 16; o >= 1; o >>= 1) mm = fmaxf(mm, __shfl_xor(mm, o, 32));
    float ssum = wave_sum32(s[t] * __expf(m[t] - mm));   // rescale to wave max
    float dsum = wave_sum32(d[t]);
    if (lane == 0) { Lm[t][wv] = mm; Ls[t][wv] = ssum; Ld[t][wv] = dsum; }
  }
  float sxw = wave_sum32(sx);
  if (lane == 0) Lsx[wv] = sxw;
  __syncthreads();

  if (tid == 0) {
    float SX = 0.f;
#pragma unroll
    for (int i = 0; i < 8; ++i) SX += Lsx[i];
#pragma unroll
    for (int t = 0; t < 3; ++t) {
      float M = -INFINITY;
#pragma unroll
      for (int i = 0; i < 8; ++i) M = fmaxf(M, Lm[t][i]);
      float S = 0.f, Dt = 0.f;
#pragma unroll
      for (int i = 0; i < 8; ++i) {
        S = fmaf(Ls[t][i], __expf(Lm[t][i] - M), S);
        Dt += Ld[t][i];
      }
      ws0[t * BB + row] = Dt - (M + __logf(S)) * SX;
    }
  }
}

// ---------------------------------------------------------------------------
// Kernel 2: KLD1, KLD2, Wasserstein element sums over [4096 x 512].
// Exactly one float4 per thread (2048 blocks * 256 threads = 524288 float4).
// Per-block partials -> ws1[q*2048 + block], q = {kld1, kld2, w}.
// ---------------------------------------------------------------------------
__global__ __launch_bounds__(256) void gauss_kernel(
    const float* __restrict__ mu, const float* __restrict__ lv,
    const float* __restrict__ pmu, const float* __restrict__ plv,
    float* __restrict__ ws1) {
  const int idx = blockIdx.x * 256 + threadIdx.x;  // float4 index, < GAUSS_N4
  float k1 = 0.f, k2 = 0.f, w = 0.f;
  if (idx < GAUSS_N4) {
    float4 m4 = reinterpret_cast<const float4*>(mu)[idx];
    float4 v4 = reinterpret_cast<const float4*>(lv)[idx];
    float4 pm4 = reinterpret_cast<const float4*>(pmu)[idx];
    float4 pv4 = reinterpret_cast<const float4*>(plv)[idx];
    float mm[4] = {m4.x, m4.y, m4.z, m4.w};
    float vv[4] = {v4.x, v4.y, v4.z, v4.w};
    float pm[4] = {pm4.x, pm4.y, pm4.z, pm4.w};
    float pv[4] = {pv4.x, pv4.y, pv4.z, pv4.w};
#pragma unroll
    for (int j = 0; j < 4; ++j) {
      float e1 = __expf(vv[j]);
      float e2 = __expf(pv[j]);
      k1 += 1.0f + vv[j] - mm[j] * mm[j] - e1;
      k2 += 1.0f + pv[j] - pm[j] * pm[j] - e2;
      float md = mm[j] - pm[j];
      float sd = sqrtf(e1) - sqrtf(e2);   // s1+s2-2*sqrt(s1*s2) == (sqrt s1 - sqrt s2)^2
      w = fmaf(md, md, w);
      w = fmaf(sd, sd, w);
    }
  }
  __shared__ float buf8[8];
  float K1 = block_sum256(k1, buf8);
  float K2 = block_sum256(k2, buf8);
  float W = block_sum256(w, buf8);
  if (threadIdx.x == 0) {
    ws1[0 * K2_BLOCKS + blockIdx.x] = K1;
    ws1[1 * K2_BLOCKS + blockIdx.x] = K2;
    ws1[2 * K2_BLOCKS + blockIdx.x] = W;
  }
}

// ---------------------------------------------------------------------------
// Kernel 3: single block. Deterministic tree reduction of all partials and
// final composition of the 6 outputs.
// ---------------------------------------------------------------------------
__global__ __launch_bounds__(256) void final_kernel(
    const float* __restrict__ ws0, const float* __restrict__ ws1,
    float* __restrict__ out) {
  const int t = threadIdx.x;
  float a0 = 0.f, a1 = 0.f, a2 = 0.f, b0 = 0.f, b1 = 0.f, b2 = 0.f;
  for (int i = t; i < BB; i += 256) {
    a0 += ws0[i];
    a1 += ws0[BB + i];
    a2 += ws0[2 * BB + i];
  }
  for (int i = t; i < K2_BLOCKS; i += 256) {
    b0 += ws1[i];
    b1 += ws1[K2_BLOCKS + i];
    b2 += ws1[2 * K2_BLOCKS + i];
  }
  __shared__ float buf8[8];
  float S0 = block_sum256(a0, buf8);   // recon_x   -> BCE_merged
  float S1 = block_sum256(a1, buf8);   // logits_rec -> BCE_rec
  float S2 = block_sum256(a2, buf8);   // logits_text -> BCE_text
  float K1 = block_sum256(b0, buf8);
  float K2 = block_sum256(b1, buf8);
  float W = block_sum256(b2, buf8);
  if (t == 0) {
    const float invBV = 1.0f / ((float)BB * (float)VV);
    const float invBD = 1.0f / ((float)BB * (float)DD);
    float bce_merged = -S0 * invBV;
    float bce_rec = -S1 * invBV;
    float bce_text = -S2 * invBV;
    float bce = (bce_merged + bce_text + bce_rec) * (1.0f / 3.0f);
    float kld1 = -0.5f * K1 * invBD;
    float kld2 = -0.5f * K2 * invBD;
    float wl = W / (float)BB;
    // ANNEAL = 1.0, EPSILON = 1.0
    float l = bce + 0.5f * (kld1 + kld2) + wl;
    out[0] = l;
    out[1] = bce;
    out[2] = wl;
    out[3] = bce_rec;
    out[4] = bce_text;
    out[5] = bce_merged;
  }
}

// ---------------------------------------------------------------------------
// Launch. Inputs (setup_inputs order):
//   0 recon_x [B,V]  1 logits_rec [B,V]  2 logits_text [B,V]  3 x [B,V]
//   4 mu [B,D]  5 logvar [B,D]  6 prior_mu [B,D]  7 prior_logvar [B,D]
// d_out: 6 f32. d_ws: ws0 = 3*4096 f32, ws1 = 3*2048 f32 (72 KB total).
// ---------------------------------------------------------------------------
extern "C" void kernel_launch(void* const* d_in, const int* in_sizes, int n_in,
                              void* d_out, int out_size, void* d_ws,
                              size_t ws_size, hipStream_t stream) {
  (void)in_sizes; (void)n_in; (void)out_size; (void)ws_size;
  const float* recon_x = (const float*)d_in[0];
  const float* logits_rec = (const float*)d_in[1];
  const float* logits_text = (const float*)d_in[2];
  const float* x = (const float*)d_in[3];
  const float* mu = (const float*)d_in[4];
  const float* logvar = (const float*)d_in[5];
  const float* prior_mu = (const float*)d_in[6];
  const float* prior_logvar = (const float*)d_in[7];

  float* ws0 = (float*)d_ws;          // 3 * 4096
  float* ws1 = ws0 + 3 * BB;          // 3 * 2048
  float* out = (float*)d_out;

  bce_row_kernel<<<dim3(BB), dim3(256), 0, stream>>>(recon_x, logits_rec,
                                                     logits_text, x, ws0);
  gauss_kernel<<<dim3(K2_BLOCKS), dim3(256), 0, stream>>>(mu, logvar, prior_mu,
                                                          prior_logvar, ws1);
  final_kernel<<<dim3(1), dim3(256), 0, stream>>>(ws0, ws1, out);
}